// ANN_Comp_29240137351521
// MI455X (gfx1250) — compile-verified
//
#include <hip/hip_runtime.h>
#include <math.h>

// Problem sizes (match reference)
#define BB 16384
#define DD 512
#define HH 512
#define CC 10
#define LL 1024          // FFT length (2H-1 = 1023 <= 1024, power of two)

typedef __attribute__((ext_vector_type(2))) float v2f;
typedef __attribute__((ext_vector_type(8))) float v8f;

// ---- CDNA5 async global->LDS path (guarded; falls back to plain staging) ----
#if defined(__has_builtin)
#if __has_builtin(__builtin_amdgcn_global_load_async_to_lds_b128) && \
    __has_builtin(__builtin_amdgcn_s_wait_asynccnt)
#define HAS_ASYNC_LDS 1
#endif
#endif

#ifdef HAS_ASYNC_LDS
typedef int v4i __attribute__((vector_size(16)));
__device__ __forceinline__ void async_copy16(const void* g, void* l) {
  // GLOBAL_LOAD_ASYNC_TO_LDS_B128: 16B global -> LDS, tracked by ASYNCcnt
  __builtin_amdgcn_global_load_async_to_lds_b128((v4i*)g, (v4i*)l, 0, 0);
}
#endif

// ---------------------------------------------------------------------------
// Kernel A: precompute w_hat[f,c] = sum_k wlast[k,c] * e^{+2*pi*i*f*k/L}
// (folds the inverse FFT + final projection into one [L x C] complex matrix)
// Exact phase via (f*k) mod 1024 -> no argument-reduction error.
// ---------------------------------------------------------------------------
__global__ __launch_bounds__(256) void dft_wlast_kernel(
    const float* __restrict__ wr, const float* __restrict__ wi,
    float2* __restrict__ wh) {
  int id = blockIdx.x * 256 + threadIdx.x;
  if (id >= LL * CC) return;
  int f = id / CC;
  int c = id - f * CC;
  float accR = 0.f, accI = 0.f;
  for (int k = 0; k < 2 * HH - 1; ++k) {
    int ph = (f * k) & (LL - 1);
    float ang = 6.283185307179586f * (float)ph * (1.0f / (float)LL);
    float sn, cs;
    __sincosf(ang, &sn, &cs);
    float a = wr[k * CC + c];
    float b = wi[k * CC + c];
    accR += a * cs - b * sn;   // (a + ib) * (cs + i sn)
    accI += a * sn + b * cs;
  }
  wh[id] = make_float2(accR, accI);
}

// ---------------------------------------------------------------------------
// Kernel B: complex GEMM h = x @ w0 via 3M (Karatsuba):
//   P1 = Xr*Wr, P2 = Xi*Wi, P3 = (Xr+Xi)*(Wr+Wi)
//   hr = P1 - P2 ; hi = P3 - P1 - P2
// Block tile: 32 (M) x 128 (N), 8 wave32s; each wave computes two 16x16
// WMMA tiles (M-split) sharing one B fragment -> fewer ds reads per WMMA.
// Panels staged via GLOBAL_LOAD_ASYNC_TO_LDS_B128 (ASYNCcnt) when available.
// Matrix path: V_WMMA_F32_16X16X4_F32 (native fp32 WMMA on CDNA5).
// ---------------------------------------------------------------------------
#define KCH 32
#define TN  128
#define TM  32

__global__ __launch_bounds__(256) void gemm3m_kernel(
    const float* __restrict__ Xr, const float* __restrict__ Xi,
    const float* __restrict__ Wr, const float* __restrict__ Wi,
    float* __restrict__ hr, float* __restrict__ hi) {
  __shared__ __align__(16) float sA[3][TM][KCH];   // 12 KB (r, i, r+i planes)
  __shared__ __align__(16) float sB[3][KCH][TN];   // 48 KB
  const int tid  = threadIdx.x;
  const int wave = tid >> 5;
  const int lane = tid & 31;
  const int row0 = blockIdx.y * TM;
  const int col0 = blockIdx.x * TN;

  v8f acc1a = {}, acc1b = {}; // Xr*Wr           (M tile 0 / 1)
  v8f acc2a = {}, acc2b = {}; // Xi*Wi
  v8f acc3a = {}, acc3b = {}; // (Xr+Xi)*(Wr+Wi)

  // f32 WMMA 16x16x4 fragment mapping (wave32):
  //   A (16x4):  lane -> M = lane&15 ; VGPR0/1 hold K = kb, kb+1, kb = 2*(lane>>4)
  //   B (4x16):  lane -> N = lane&15 ; VGPR0/1 hold K = kb, kb+1
  const int m_a = lane & 15;
  const int kb  = (lane >> 4) << 1;
  const int n_b = lane & 15;
  const int n0  = wave << 4;

  for (int k0 = 0; k0 < DD; k0 += KCH) {
#ifdef HAS_ASYNC_LDS
    // --- async global->LDS staging, 16B chunks ---
    {
      // A planes: 32x32 floats = 256 chunks per plane, 1 chunk/thread
      int m  = tid >> 3;
      int n4 = (tid & 7) << 2;
      async_copy16(&Xr[(size_t)(row0 + m) * DD + k0 + n4], &sA[0][m][n4]);
      async_copy16(&Xi[(size_t)(row0 + m) * DD + k0 + n4], &sA[1][m][n4]);
    }
#pragma unroll
    for (int j = 0; j < 4; ++j) {
      // B planes: 32x128 floats = 1024 chunks per plane, 4 chunks/thread
      int c  = tid + j * 256;
      int k  = c >> 5;
      int n4 = (c & 31) << 2;
      async_copy16(&Wr[(size_t)(k0 + k) * HH + col0 + n4], &sB[0][k][n4]);
      async_copy16(&Wi[(size_t)(k0 + k) * HH + col0 + n4], &sB[1][k][n4]);
    }
    __builtin_amdgcn_s_wait_asynccnt(0);
    __syncthreads();
    // Build Karatsuba sum planes in LDS
    for (int idx = tid; idx < TM * KCH; idx += 256) {
      int m = idx >> 5, k = idx & (KCH - 1);
      sA[2][m][k] = sA[0][m][k] + sA[1][m][k];
    }
    for (int idx = tid; idx < KCH * TN; idx += 256) {
      int k = idx >> 7, n = idx & (TN - 1);
      sB[2][k][n] = sB[0][k][n] + sB[1][k][n];
    }
    __syncthreads();
#else
    // --- fallback: plain staged loads, sums computed inline ---
    for (int idx = tid; idx < TM * KCH; idx += 256) {
      int m = idx >> 5, k = idx & (KCH - 1);
      float xr = Xr[(size_t)(row0 + m) * DD + k0 + k];
      float xi = Xi[(size_t)(row0 + m) * DD + k0 + k];
      sA[0][m][k] = xr;
      sA[1][m][k] = xi;
      sA[2][m][k] = xr + xi;
    }
    for (int idx = tid; idx < KCH * TN; idx += 256) {
      int k = idx >> 7, n = idx & (TN - 1);
      float wr = Wr[(size_t)(k0 + k) * HH + col0 + n];
      float wi = Wi[(size_t)(k0 + k) * HH + col0 + n];
      sB[0][k][n] = wr;
      sB[1][k][n] = wi;
      sB[2][k][n] = wr + wi;
    }
    __syncthreads();
    if (k0 + KCH < DD) {
      int pk = k0 + KCH + (tid >> 3);
      int pn = col0 + ((tid & 7) << 4);
      __builtin_prefetch(&Wr[(size_t)pk * HH + pn], 0, 1);
      __builtin_prefetch(&Wi[(size_t)pk * HH + pn], 0, 1);
    }
#endif

#pragma unroll
    for (int kk = 0; kk < KCH; kk += 4) {
      v2f br, bi, bs;
      br.x = sB[0][kk + kb][n0 + n_b]; br.y = sB[0][kk + kb + 1][n0 + n_b];
      bi.x = sB[1][kk + kb][n0 + n_b]; bi.y = sB[1][kk + kb + 1][n0 + n_b];
      bs.x = sB[2][kk + kb][n0 + n_b]; bs.y = sB[2][kk + kb + 1][n0 + n_b];

      v2f ar0, ai0, as0, ar1, ai1, as1;
      ar0.x = sA[0][m_a][kk + kb];      ar0.y = sA[0][m_a][kk + kb + 1];
      ai0.x = sA[1][m_a][kk + kb];      ai0.y = sA[1][m_a][kk + kb + 1];
      as0.x = sA[2][m_a][kk + kb];      as0.y = sA[2][m_a][kk + kb + 1];
      ar1.x = sA[0][16 + m_a][kk + kb]; ar1.y = sA[0][16 + m_a][kk + kb + 1];
      ai1.x = sA[1][16 + m_a][kk + kb]; ai1.y = sA[1][16 + m_a][kk + kb + 1];
      as1.x = sA[2][16 + m_a][kk + kb]; as1.y = sA[2][16 + m_a][kk + kb + 1];

      acc1a = __builtin_amdgcn_wmma_f32_16x16x4_f32(false, ar0, false, br,
                                                    (short)0, acc1a, false, false);
      acc1b = __builtin_amdgcn_wmma_f32_16x16x4_f32(false, ar1, false, br,
                                                    (short)0, acc1b, false, false);
      acc2a = __builtin_amdgcn_wmma_f32_16x16x4_f32(false, ai0, false, bi,
                                                    (short)0, acc2a, false, false);
      acc2b = __builtin_amdgcn_wmma_f32_16x16x4_f32(false, ai1, false, bi,
                                                    (short)0, acc2b, false, false);
      acc3a = __builtin_amdgcn_wmma_f32_16x16x4_f32(false, as0, false, bs,
                                                    (short)0, acc3a, false, false);
      acc3b = __builtin_amdgcn_wmma_f32_16x16x4_f32(false, as1, false, bs,
                                                    (short)0, acc3b, false, false);
    }
    __syncthreads();
  }

  // Epilogue: C/D layout — VGPR i: lanes 0-15 -> M=i, lanes 16-31 -> M=8+i
  const int mbase = (lane >> 4) << 3;
  const int col   = col0 + n0 + n_b;
#pragma unroll
  for (int i = 0; i < 8; ++i) {
    int r0 = row0 + mbase + i;        // M tile 0
    int r1 = row0 + 16 + mbase + i;   // M tile 1
    hr[(size_t)r0 * HH + col] = acc1a[i] - acc2a[i];
    hi[(size_t)r0 * HH + col] = acc3a[i] - acc1a[i] - acc2a[i];
    hr[(size_t)r1 * HH + col] = acc1b[i] - acc2b[i];
    hi[(size_t)r1 * HH + col] = acc3b[i] - acc1b[i] - acc2b[i];
  }
}

// ---------------------------------------------------------------------------
// Kernel C: per row -- 1024-pt radix-2 DIT FFT in LDS, pointwise square
// (conv theorem), project against w_hat (folded IFFT + wlast), magnitude.
// One block (256 threads = 8 waves) per row.
// ---------------------------------------------------------------------------
__global__ __launch_bounds__(256) void fft_square_proj_kernel(
    const float* __restrict__ hr, const float* __restrict__ hi,
    const float2* __restrict__ wh, float* __restrict__ out) {
  __shared__ float2 sh[LL];        //  8 KB
  __shared__ float2 red[CC][256];  // 20 KB
  const int row = blockIdx.x;
  const int tid = threadIdx.x;

  // zero-pad, then bit-reversed scatter of the 512 input samples
  for (int f = tid; f < LL; f += 256) sh[f] = make_float2(0.f, 0.f);
  __syncthreads();
  for (int n = tid; n < HH; n += 256) {
    int rev = (int)(__brev((unsigned)n) >> 22); // 10-bit reversal
    sh[rev] = make_float2(hr[(size_t)row * HH + n], hi[(size_t)row * HH + n]);
  }
  __syncthreads();

  // 10 butterfly stages, forward transform (e^{-i...})
  for (int s = 1; s <= 10; ++s) {
    int m = 1 << s, mh = m >> 1;
    for (int j = tid; j < (LL >> 1); j += 256) {
      int grp = j / mh;
      int pos = j - grp * mh;
      int i0 = grp * m + pos;
      int i1 = i0 + mh;
      float ang = -6.283185307179586f * (float)pos / (float)m;
      float sn, cs;
      __sincosf(ang, &sn, &cs);
      float2 u = sh[i0];
      float2 v = sh[i1];
      float tr = v.x * cs - v.y * sn;
      float ti = v.x * sn + v.y * cs;
      sh[i0] = make_float2(u.x + tr, u.y + ti);
      sh[i1] = make_float2(u.x - tr, u.y - ti);
    }
    __syncthreads();
  }

  // pointwise square: s = hhat^2
  for (int f = tid; f < LL; f += 256) {
    float2 v = sh[f];
    sh[f] = make_float2(v.x * v.x - v.y * v.y, 2.f * v.x * v.y);
  }
  __syncthreads();

  // projection: acc[c] = sum_f s[f] * w_hat[f,c]
  float accR[CC] = {}, accI[CC] = {};
  for (int f = tid; f < LL; f += 256) {
    float2 s = sh[f];
#pragma unroll
    for (int c = 0; c < CC; ++c) {
      float2 w = wh[f * CC + c];
      accR[c] += s.x * w.x - s.y * w.y;
      accI[c] += s.x * w.y + s.y * w.x;
    }
  }
#pragma unroll
  for (int c = 0; c < CC; ++c) red[c][tid] = make_float2(accR[c], accI[c]);
  __syncthreads();
  for (int st = 128; st > 0; st >>= 1) {
    if (tid < st) {
#pragma unroll
      for (int c = 0; c < CC; ++c) {
        float2 a = red[c][tid], b = red[c][tid + st];
        red[c][tid] = make_float2(a.x + b.x, a.y + b.y);
      }
    }
    __syncthreads();
  }
  if (tid < CC) {
    float2 r = red[tid][0];
    const float sc = 1.0f / (float)LL; // IFFT scale
    float re = r.x * sc, im = r.y * sc;
    out[(size_t)row * CC + tid] = sqrtf(re * re + im * im);
  }
}

// ---------------------------------------------------------------------------
// Launch
// ---------------------------------------------------------------------------
extern "C" void kernel_launch(void* const* d_in, const int* in_sizes, int n_in,
                              void* d_out, int out_size, void* d_ws, size_t ws_size,
                              hipStream_t stream) {
  (void)in_sizes; (void)n_in; (void)out_size; (void)ws_size;
  const float* xr  = (const float*)d_in[0];
  const float* xi  = (const float*)d_in[1];
  const float* w0r = (const float*)d_in[2];
  const float* w0i = (const float*)d_in[3];
  const float* wlr = (const float*)d_in[4];
  const float* wli = (const float*)d_in[5];

  char* ws = (char*)d_ws;
  float2* wh = (float2*)ws;                                   // 1024*10 complex
  size_t wh_bytes = ((size_t)LL * CC * sizeof(float2) + 255) & ~(size_t)255;
  float* hr = (float*)(ws + wh_bytes);                        // B*H floats
  float* hi = hr + (size_t)BB * HH;                           // B*H floats

  dft_wlast_kernel<<<(LL * CC + 255) / 256, 256, 0, stream>>>(wlr, wli, wh);
  gemm3m_kernel<<<dim3(HH / TN, BB / TM), 256, 0, stream>>>(xr, xi, w0r, w0i, hr, hi);
  fft_square_proj_kernel<<<BB, 256, 0, stream>>>(hr, hi, wh, (float*)d_out);
}